// CorrelationSimilarity_53867479827090
// MI455X (gfx1250) — compile-verified
//
#include <hip/hip_runtime.h>
#include <hip/hip_bf16.h>
#include <math.h>

// CorrelationSimilarity on gfx1250:
//   g[b,n,u,v] = sum_c sum_{h,w} x[b,c,h,w] * y[n,c,(h+u)%64,(w+v)%64]
//   out[b,n]   = max_{u,v} g / (||x_b|| * ||y_n||)
// via DFT-as-matmul on V_WMMA_F32_16X16X4_F32 (full fp32 precision).

typedef __attribute__((ext_vector_type(2))) float v2f;
typedef __attribute__((ext_vector_type(8))) float v8f;

#define LSTR 65            // LDS row stride in floats (bank-conflict padding)
#define PLANE (64 * LSTR)  // one 64x64 plane
#define PAIRS_PER_WG 4

// D = s1*(A1@B1) + s2*(A2@B2); 64x64 operands in LDS.
// A1/A2 row-major [m][k]; B1t/B2t TRANSPOSED, i.e. Bt[n][k] = B[k][n]
// (so both A and B fragment element pairs are LDS-adjacent -> ds_load_2addr).
// Block = 128 threads = 4 waves; wave w owns output rows [16w, 16w+16).
// D written row-major.
__device__ __forceinline__ void wave_mm64(const float* A1, const float* B1t, float s1,
                                          const float* A2, const float* B2t, float s2,
                                          float* D) {
  const int lane = threadIdx.x & 31;
  const int m16  = (threadIdx.x >> 5) << 4;
  const int row  = m16 + (lane & 15);
  const int koff = (lane >> 4) << 1;  // lanes 16-31 take K+2,K+3

  // Preload all A fragments for this row strip (reused across 4 n-tiles);
  // fold product signs into A.
  v2f a1[16], a2[16];
#pragma unroll
  for (int kk = 0; kk < 16; ++kk) {
    int kb = 4 * kk + koff;
    v2f a;
    a.x = s1 * A1[row * LSTR + kb];
    a.y = s1 * A1[row * LSTR + kb + 1];
    a1[kk] = a;
  }
  if (A2) {
#pragma unroll
    for (int kk = 0; kk < 16; ++kk) {
      int kb = 4 * kk + koff;
      v2f a;
      a.x = s2 * A2[row * LSTR + kb];
      a.y = s2 * A2[row * LSTR + kb + 1];
      a2[kk] = a;
    }
  }

#pragma unroll
  for (int n16 = 0; n16 < 64; n16 += 16) {
    const int col = n16 + (lane & 15);
    v8f acc = {};
#pragma unroll
    for (int kk = 0; kk < 16; ++kk) {
      int kb = 4 * kk + koff;
      v2f b;
      b.x = B1t[col * LSTR + kb];
      b.y = B1t[col * LSTR + kb + 1];
      acc = __builtin_amdgcn_wmma_f32_16x16x4_f32(false, a1[kk], false, b, (short)0,
                                                  acc, false, false);
    }
    if (A2) {
#pragma unroll
      for (int kk = 0; kk < 16; ++kk) {
        int kb = 4 * kk + koff;
        v2f b;
        b.x = B2t[col * LSTR + kb];
        b.y = B2t[col * LSTR + kb + 1];
        acc = __builtin_amdgcn_wmma_f32_16x16x4_f32(false, a2[kk], false, b, (short)0,
                                                    acc, false, false);
      }
    }
    // C/D layout: VGPR r -> M = m16 + r + 8*(lane>=16), N = n16 + (lane&15)
    const int rbase = m16 + ((lane >> 4) << 3);
#pragma unroll
    for (int r = 0; r < 8; ++r) D[(rbase + r) * LSTR + col] = acc[r];
  }
}

// cs[k][t] = cos(2*pi*k*t/64), sn[k][t] = sin(2*pi*k*t/64); both symmetric.
__device__ __forceinline__ void fill_tables(float* cs, float* sn) {
  const float w0 = 6.28318530717958647692f / 64.0f;
  for (int idx = threadIdx.x; idx < 4096; idx += blockDim.x) {
    int k = idx >> 6, t = idx & 63;
    float ang = w0 * (float)((k * t) & 63);
    float s, c;
    __sincosf(ang, &s, &c);
    cs[k * LSTR + t] = c;
    sn[k * LSTR + t] = s;
  }
}

// ---------------- Kernel 1: L2 norms of x rows (32) and y rows (512) --------
__global__ void norms_kernel(const float* __restrict__ x, const float* __restrict__ y,
                             float* __restrict__ norms) {
  __shared__ float red[256];
  int j = blockIdx.x;  // 0..543
  const float* base = (j < 32) ? (x + (size_t)j * 12288)
                               : (y + (size_t)(j - 32) * 12288);
  float s = 0.f;
  for (int i = threadIdx.x; i < 12288; i += blockDim.x) {
    float v = base[i];
    s += v * v;
  }
  red[threadIdx.x] = s;
  __syncthreads();
  for (int off = 128; off > 0; off >>= 1) {
    if ((int)threadIdx.x < off) red[threadIdx.x] += red[threadIdx.x + off];
    __syncthreads();
  }
  if (threadIdx.x == 0) norms[j] = sqrtf(red[0]);
}

// ---------------- Kernel 2: forward 2D DFT (Z = W z W, W = cs - i*sn) -------
// One workgroup (128 threads) per image; 1632 images (96 from x, 1536 from y).
__global__ void fwd_dft_kernel(const float* __restrict__ x, const float* __restrict__ y,
                               float* __restrict__ spec_r, float* __restrict__ spec_i) {
  extern __shared__ float smem[];
  float* cs = smem;         // symmetric: serves as A or Bt
  float* sn = cs + PLANE;   // symmetric
  float* zT = sn + PLANE;   // input image, TRANSPOSED (B operand only)
  float* Sr = zT + PLANE;   // A operand only
  float* P  = Sr + PLANE;   // P = sn@z  (Si = -P); A operand only
  float* Zr = P  + PLANE;
  float* Zi = Zr + PLANE;

  int img = blockIdx.x;  // 0..1631
  const float* src = (img < 96) ? (x + (size_t)img * 4096)
                                : (y + (size_t)(img - 96) * 4096);
  fill_tables(cs, sn);
  for (int idx = threadIdx.x; idx < 4096; idx += blockDim.x) {
    int h = idx >> 6, w = idx & 63;
    zT[w * LSTR + h] = src[idx];  // transposed store
  }
  __syncthreads();

  wave_mm64(cs, zT, 1.f, nullptr, nullptr, 0.f, Sr);  // Sr =  cs@z
  wave_mm64(sn, zT, 1.f, nullptr, nullptr, 0.f, P);   // P  =  sn@z
  __syncthreads();
  wave_mm64(Sr, cs, 1.f, P, sn, -1.f, Zr);            // Zr =  Sr@cs - P@sn
  wave_mm64(P, cs, -1.f, Sr, sn, -1.f, Zi);           // Zi = -P@cs  - Sr@sn
  __syncthreads();

  float* outr = spec_r + (size_t)img * 4096;
  float* outi = spec_i + (size_t)img * 4096;
  for (int idx = threadIdx.x; idx < 4096; idx += blockDim.x) {
    int h = idx >> 6, w = idx & 63;
    outr[idx] = Zr[h * LSTR + w];
    outi[idx] = Zi[h * LSTR + w];
  }
}

// ---------------- Kernel 3: per-(b,n) cross-spectrum, inverse DFT, max ------
// raw[u,v] = Re( Wc @ G @ Wc ), Wc = cs + i*sn; g = raw/4096.
// Each workgroup handles PAIRS_PER_WG consecutive n for one b (tables built once).
__global__ void corr_kernel(const float* __restrict__ spec_r,
                            const float* __restrict__ spec_i,
                            const float* __restrict__ norms,
                            float* __restrict__ out) {
  extern __shared__ float smem[];
  float* cs  = smem;
  float* sn  = cs + PLANE;
  float* GrT = sn + PLANE;   // cross-spectrum, TRANSPOSED (B operand only)
  float* GiT = GrT + PLANE;  // TRANSPOSED (B operand only)
  float* Tr  = GiT + PLANE;  // A operand only
  float* Ti  = Tr + PLANE;   // A operand only
  float* g   = Ti + PLANE;
  float* red = g + PLANE;    // 128 floats of reduction scratch

  const int pair0 = blockIdx.x * PAIRS_PER_WG;  // b*512 + n0
  const int b  = pair0 >> 9;
  const int n0 = pair0 & 511;

  fill_tables(cs, sn);

  const float* Xr0 = spec_r + (size_t)(b * 3) * 4096;
  const float* Xi0 = spec_i + (size_t)(b * 3) * 4096;
  const float xnorm = norms[b];

  for (int p = 0; p < PAIRS_PER_WG; ++p) {
    const int n = n0 + p;
    const float* Yr0 = spec_r + (size_t)(96 + n * 3) * 4096;
    const float* Yi0 = spec_i + (size_t)(96 + n * 3) * 4096;

    __syncthreads();  // previous iteration fully done with GrT/GiT/red
    // G = sum_c conj(X[b,c]) * Y[n,c], stored transposed
    for (int idx = threadIdx.x; idx < 4096; idx += blockDim.x) {
      float gr = 0.f, gi = 0.f;
#pragma unroll
      for (int c = 0; c < 3; ++c) {
        float xr = Xr0[c * 4096 + idx], xi = Xi0[c * 4096 + idx];
        float yr = Yr0[c * 4096 + idx], yi = Yi0[c * 4096 + idx];
        gr += xr * yr + xi * yi;  // Re(conj(X)*Y)
        gi += xr * yi - xi * yr;  // Im(conj(X)*Y)
      }
      int h = idx >> 6, w = idx & 63;
      GrT[w * LSTR + h] = gr;
      GiT[w * LSTR + h] = gi;
    }
    __syncthreads();

    wave_mm64(cs, GrT, 1.f, sn, GiT, -1.f, Tr);  // Tr = cs@Gr - sn@Gi
    wave_mm64(cs, GiT, 1.f, sn, GrT, 1.f, Ti);   // Ti = cs@Gi + sn@Gr
    __syncthreads();
    wave_mm64(Tr, cs, 1.f, Ti, sn, -1.f, g);     // raw = Tr@cs - Ti@sn
    __syncthreads();

    float m = -3.402823466e38f;
    for (int idx = threadIdx.x; idx < 4096; idx += blockDim.x) {
      int h = idx >> 6, w = idx & 63;
      m = fmaxf(m, g[h * LSTR + w]);
    }
    red[threadIdx.x] = m;
    __syncthreads();
    for (int off = 64; off > 0; off >>= 1) {
      if ((int)threadIdx.x < off)
        red[threadIdx.x] = fmaxf(red[threadIdx.x], red[threadIdx.x + off]);
      __syncthreads();
    }
    if (threadIdx.x == 0)
      out[pair0 + p] = red[0] / (4096.0f * xnorm * norms[32 + n]);
  }
}

extern "C" void kernel_launch(void* const* d_in, const int* in_sizes, int n_in,
                              void* d_out, int out_size, void* d_ws, size_t ws_size,
                              hipStream_t stream) {
  (void)in_sizes; (void)n_in; (void)out_size; (void)ws_size;
  const float* x = (const float*)d_in[0];  // (32,3,64,64)
  const float* y = (const float*)d_in[1];  // (512,3,64,64)
  float* out = (float*)d_out;              // (32,512)

  // Workspace: planar spectra (x images 0..95, y images 96..1631), then norms.
  float* spec_r = (float*)d_ws;                   // 1632*4096 floats
  float* spec_i = spec_r + (size_t)1632 * 4096;   // 1632*4096 floats
  float* norms  = spec_i + (size_t)1632 * 4096;   // 544 floats

  size_t shmem = ((size_t)7 * PLANE + 128) * sizeof(float);  // ~117 KB/WG

  hipLaunchKernelGGL(norms_kernel, dim3(544), dim3(256), 0, stream, x, y, norms);
  hipLaunchKernelGGL(fwd_dft_kernel, dim3(1632), dim3(128), shmem, stream,
                     x, y, spec_r, spec_i);
  hipLaunchKernelGGL(corr_kernel, dim3((32 * 512) / PAIRS_PER_WG), dim3(128), shmem,
                     stream, spec_r, spec_i, norms, out);
}